// IntegratedBiLSTMModel_66623532696225
// MI455X (gfx1250) — compile-verified
//
#include <hip/hip_runtime.h>
#include <math.h>

// ---------------------------------------------------------------------------
// Bidirectional LSTM for MI455X (gfx1250, wave32, WMMA).
//   B=32, S=512, I=H=512.
// Phase 1: Gx[dir][t][b][4H] = X @ Wih_dir^T + (bih+bhh)   (big f32 WMMA GEMM)
// Phase 2: persistent recurrent scan, 32 WGs/direction, hidden-unit
//          partitioned so cell state c stays WG-local in LDS; per-step
//          32x64x512 f32 WMMA GEMM with Whh slice resident in LDS; global
//          atomic grid barrier per direction per step.
// ---------------------------------------------------------------------------

typedef __attribute__((ext_vector_type(2))) float v2f;
typedef __attribute__((ext_vector_type(8))) float v8f;
typedef __attribute__((ext_vector_type(4))) float f4;

#define BATCH 32
#define SEQ   512
#define HID   512
#define G4H   2048

// workspace layout (floats)
#define WS_HBUF 64                       // counters occupy [0,64)
#define HBUF_SZ (2 * 2 * BATCH * HID)    // 2 dir * 2 slots * 32*512 = 65536
#define WS_GXF  (WS_HBUF + HBUF_SZ)      // 65600
#define GX_SZ   ((size_t)SEQ * BATCH * G4H)  // 33,554,432 floats per dir
#define WS_GXB  (WS_GXF + GX_SZ)

#define WSTR 516   // LDS row pad: 516 mod 64 banks = 4 -> conflict-free b64
#define GSTR 68

#define SMEM_B_BYTES ((64 * WSTR + 32 * WSTR + 32 * GSTR + 512) * 4)  // 208896

// ---------------------------------------------------------------------------
// init: zero barrier counters + h double buffers (d_ws is poisoned 0xAA and
// NOT re-poisoned between graph replays, so this runs every call).
// ---------------------------------------------------------------------------
__global__ void bilstm_init(float* ws) {
  int i = blockIdx.x * blockDim.x + threadIdx.x;
  if (i < WS_HBUF + HBUF_SZ) ws[i] = 0.0f;
}

// ---------------------------------------------------------------------------
// Phase 1: Gx = X @ Wih^T + bias, stored time-major [t][b][4H].
// grid (512, 32, 2): 32(M) x 64(N) macro tile, 8 waves, 1 16x16 tile/wave.
// ---------------------------------------------------------------------------
__global__ __launch_bounds__(256)
void bilstm_gatex_gemm(const float* __restrict__ X,
                       const float* __restrict__ Wih_f,
                       const float* __restrict__ bih_f,
                       const float* __restrict__ bhh_f,
                       const float* __restrict__ Wih_b,
                       const float* __restrict__ bih_b,
                       const float* __restrict__ bhh_b,
                       float* __restrict__ Gx_f,
                       float* __restrict__ Gx_b) {
  const int dir = blockIdx.z;
  const float* W  = dir ? Wih_b : Wih_f;
  const float* bA = dir ? bih_b : bih_f;
  const float* bB = dir ? bhh_b : bhh_f;
  float* Gx = dir ? Gx_b : Gx_f;

  __shared__ float Xs[32 * GSTR];   // 32 rows x 64 K (pad 68)
  __shared__ float Ws[64 * GSTR];   // 64 rows x 64 K (pad 68)

  const int tid  = threadIdx.x;
  const int wave = tid >> 5, lane = tid & 31;
  const int mi = wave >> 2, nj = wave & 3;    // 2(M) x 4(N) wave grid
  const int ln = lane & 15, hi = lane >> 4;
  const int m0 = blockIdx.x * 32;
  const int n0 = blockIdx.y * 64;

  v8f acc0 = {}; v8f acc1 = {};

  for (int k0 = 0; k0 < 512; k0 += 64) {
    {   // stage X tile: 2048 floats, 8/thread (2x float4), coalesced
      int r = tid >> 3, seg = tid & 7;
      const f4* src = (const f4*)(X + (size_t)(m0 + r) * 512 + k0 + seg * 8);
      f4* dst = (f4*)(Xs + r * GSTR + seg * 8);
      dst[0] = src[0]; dst[1] = src[1];
    }
    {   // stage Wih tile: 4096 floats, 16/thread (4x float4), coalesced
      int r = tid >> 2, seg = tid & 3;
      const f4* src = (const f4*)(W + (size_t)(n0 + r) * 512 + k0 + seg * 16);
      f4* dst = (f4*)(Ws + r * GSTR + seg * 16);
      dst[0] = src[0]; dst[1] = src[1]; dst[2] = src[2]; dst[3] = src[3];
    }
    __syncthreads();
    // A 16x4 frag: lanes 0-15 hold {K=k,k+1}, lanes 16-31 hold {k+2,k+3}
    const float* arow = Xs + (mi * 16 + ln) * GSTR + 2 * hi;
    const float* brow = Ws + (nj * 16 + ln) * GSTR + 2 * hi;
#pragma unroll
    for (int k = 0; k < 64; k += 8) {
      v2f a0 = *(const v2f*)(arow + k);
      v2f b0 = *(const v2f*)(brow + k);
      acc0 = __builtin_amdgcn_wmma_f32_16x16x4_f32(false, a0, false, b0,
                                                   (short)0, acc0, false, false);
      v2f a1 = *(const v2f*)(arow + k + 4);
      v2f b1 = *(const v2f*)(brow + k + 4);
      acc1 = __builtin_amdgcn_wmma_f32_16x16x4_f32(false, a1, false, b1,
                                                   (short)0, acc1, false, false);
    }
    __syncthreads();
  }
  v8f acc = acc0 + acc1;

  const int n = n0 + nj * 16 + ln;
  const float bias = bA[n] + bB[n];
#pragma unroll
  for (int r = 0; r < 8; ++r) {
    int m  = m0 + mi * 16 + r + 8 * hi;  // C/D layout: M = r (+8 upper half)
    int bb = m >> 9;                     // batch (X rows are b*S+s)
    int ss = m & 511;                    // time
    Gx[((size_t)ss * BATCH + bb) * G4H + n] = acc[r] + bias;
  }
}

// ---------------------------------------------------------------------------
// Phase 2: recurrent scan. grid = 64 WGs: dir = blk>>5, hidden group = blk&31.
// ---------------------------------------------------------------------------
__global__ __launch_bounds__(256)
void bilstm_recurrent(const float* __restrict__ Whh_f,
                      const float* __restrict__ Whh_b,
                      const float* __restrict__ Gx_f,
                      const float* __restrict__ Gx_b,
                      float* __restrict__ hbuf,
                      unsigned* __restrict__ counters,
                      float* __restrict__ out) {
  extern __shared__ float smem[];
  float* Wlds = smem;                    // 64 x 516 : Whh slice (resident)
  float* hlds = Wlds + 64 * WSTR;        // 32 x 516 : staged h_prev
  float* glds = hlds + 32 * WSTR;        // 32 x 68  : gate pre-activations
  float* clds = glds + 32 * GSTR;        // 512      : WG-private cell state

  const int tid  = threadIdx.x;
  const int dir  = blockIdx.x >> 5;
  const int grp  = blockIdx.x & 31;
  const int j0   = grp * 16;             // owned hidden units [j0, j0+16)
  const int wave = tid >> 5, lane = tid & 31;
  const int mi = wave >> 2, nj = wave & 3;   // nj = gate type (i,f,g,o)
  const int ln = lane & 15, hi = lane >> 4;

  const float* Whh = dir ? Whh_b : Whh_f;
  const float* Gx  = dir ? Gx_b : Gx_f;
  float*    hb  = hbuf + (size_t)dir * 2 * BATCH * HID;
  unsigned* ctr = counters + dir * 32;   // 128B apart per direction

  {  // preload Whh slice: row r -> Whh[(r>>4)*512 + j0 + (r&15)]
    int r = tid >> 2, seg = tid & 3;
    int ng = (r >> 4) * HID + j0 + (r & 15);
    const f4* src = (const f4*)(Whh + (size_t)ng * HID + seg * 128);
    f4* dst = (f4*)(Wlds + r * WSTR + seg * 128);
#pragma unroll
    for (int i = 0; i < 32; ++i) dst[i] = src[i];
  }
  clds[tid] = 0.0f; clds[tid + 256] = 0.0f;   // c0 = 0
  __syncthreads();

  for (int it = 0; it < SEQ; ++it) {
    const int t = dir ? (SEQ - 1 - it) : it;
    const int cur = it & 1, nxt = cur ^ 1;

    {  // stage h_prev [32][512] (slot 0 zeroed by init kernel => h0 = 0)
      int row = tid >> 3, seg = tid & 7;
      const f4* src = (const f4*)(hb + cur * BATCH * HID + row * HID + seg * 64);
      f4* dst = (f4*)(hlds + row * WSTR + seg * 64);
#pragma unroll
      for (int i = 0; i < 16; ++i) dst[i] = src[i];
    }
    __syncthreads();

    // 32x64x512 GEMM: h_prev @ Whh_slice^T ; dual accumulators break RAW chain
    v8f acc0 = {}; v8f acc1 = {};
    const float* arow = hlds + (mi * 16 + ln) * WSTR + 2 * hi;
    const float* brow = Wlds + (nj * 16 + ln) * WSTR + 2 * hi;
#pragma unroll 8
    for (int k = 0; k < 512; k += 8) {
      v2f a0 = *(const v2f*)(arow + k);
      v2f b0 = *(const v2f*)(brow + k);
      acc0 = __builtin_amdgcn_wmma_f32_16x16x4_f32(false, a0, false, b0,
                                                   (short)0, acc0, false, false);
      v2f a1 = *(const v2f*)(arow + k + 4);
      v2f b1 = *(const v2f*)(brow + k + 4);
      acc1 = __builtin_amdgcn_wmma_f32_16x16x4_f32(false, a1, false, b1,
                                                   (short)0, acc1, false, false);
    }
    v8f acc = acc0 + acc1;
#pragma unroll
    for (int r = 0; r < 8; ++r) {
      int row = mi * 16 + r + 8 * hi;
      glds[row * GSTR + nj * 16 + ln] = acc[r];
    }
    __syncthreads();

    // gate nonlinearity + state update: 512 (b,j) cells, 2 per thread
#pragma unroll
    for (int p = 0; p < 2; ++p) {
      int e = tid + p * 256;
      int bb = e >> 4, jj = e & 15;
      size_t gxb = ((size_t)t * BATCH + bb) * G4H + j0 + jj;
      float gi = glds[bb * GSTR + jj]      + Gx[gxb];
      float gf = glds[bb * GSTR + 16 + jj] + Gx[gxb + 512];
      float gg = glds[bb * GSTR + 32 + jj] + Gx[gxb + 1024];
      float go = glds[bb * GSTR + 48 + jj] + Gx[gxb + 1536];
      float si = 1.0f / (1.0f + __expf(-gi));
      float sf = 1.0f / (1.0f + __expf(-gf));
      float so = 1.0f / (1.0f + __expf(-go));
      float c  = sf * clds[e] + si * tanhf(gg);
      float h  = so * tanhf(c);
      clds[e] = c;
      hb[nxt * BATCH * HID + bb * HID + j0 + jj] = h;
      // ht: [B][S][2H]
      out[(size_t)bb * (SEQ * 2 * HID) + (size_t)t * (2 * HID) + dir * HID + j0 + jj] = h;
      if (it == SEQ - 1) {  // final (hN, cN): [2][B][H] each
        size_t base_h = (size_t)BATCH * SEQ * 2 * HID;
        size_t base_c = base_h + (size_t)2 * BATCH * HID;
        out[base_h + (size_t)dir * BATCH * HID + bb * HID + j0 + jj] = h;
        out[base_c + (size_t)dir * BATCH * HID + bb * HID + j0 + jj] = c;
      } else {  // pull next step's staged gates toward L2/L0
        int tn = dir ? (t - 1) : (t + 1);
        __builtin_prefetch(Gx + ((size_t)tn * BATCH + bb) * G4H + j0 + jj, 0, 1);
      }
    }

    // per-direction grid barrier (monotone target; counters reset each call)
    __threadfence();
    __syncthreads();
    if (tid == 0) {
      __hip_atomic_fetch_add(ctr, 1u, __ATOMIC_ACQ_REL, __HIP_MEMORY_SCOPE_AGENT);
      unsigned target = (unsigned)(it + 1) * 32u;
      while (__hip_atomic_load(ctr, __ATOMIC_ACQUIRE, __HIP_MEMORY_SCOPE_AGENT) < target) {
        __builtin_amdgcn_s_sleep(4);
      }
    }
    __syncthreads();
    __threadfence();
  }
}

// ---------------------------------------------------------------------------
extern "C" void kernel_launch(void* const* d_in, const int* in_sizes, int n_in,
                              void* d_out, int out_size, void* d_ws, size_t ws_size,
                              hipStream_t stream) {
  (void)in_sizes; (void)n_in; (void)out_size; (void)ws_size;
  const float* X     = (const float*)d_in[0];
  const float* Wih_f = (const float*)d_in[1];
  const float* Whh_f = (const float*)d_in[2];
  const float* bih_f = (const float*)d_in[3];
  const float* bhh_f = (const float*)d_in[4];
  const float* Wih_b = (const float*)d_in[5];
  const float* Whh_b = (const float*)d_in[6];
  const float* bih_b = (const float*)d_in[7];
  const float* bhh_b = (const float*)d_in[8];

  float* ws = (float*)d_ws;          // needs ~269 MB of scratch
  unsigned* ctr  = (unsigned*)d_ws;  // counters at ws[0..63]
  float* hbuf = ws + WS_HBUF;
  float* Gx_f = ws + WS_GXF;
  float* Gx_b = ws + WS_GXB;
  float* out  = (float*)d_out;

  hipFuncSetAttribute(reinterpret_cast<const void*>(&bilstm_recurrent),
                      hipFuncAttributeMaxDynamicSharedMemorySize, SMEM_B_BYTES);

  bilstm_init<<<(WS_HBUF + HBUF_SZ + 255) / 256, 256, 0, stream>>>(ws);

  dim3 gA(SEQ * BATCH / 32, G4H / 64, 2);  // (512, 32, 2)
  bilstm_gatex_gemm<<<gA, 256, 0, stream>>>(X, Wih_f, bih_f, bhh_f,
                                            Wih_b, bih_b, bhh_b, Gx_f, Gx_b);

  bilstm_recurrent<<<64, 256, SMEM_B_BYTES, stream>>>(Whh_f, Whh_b, Gx_f, Gx_b,
                                                      hbuf, ctr, out);
}